// MultiHeadEMA_5420248727969
// MI455X (gfx1250) — compile-verified
//
#include <hip/hip_runtime.h>

// MultiHeadEMA on gfx1250: chunked (C=16) state-space scan, all dense phases
// as 16x16 f32 WMMA; X tiles staged global->LDS via the Tensor Data Mover
// (tensor_load_to_lds) with hardware padding for bank-conflict-free reads.
//
//   out[b,t,d] = sum_{s<=t} k[d,t-s] x[b,s,d] + omega[d] x[b,t,d]
//   k[d,j] = sum_n w_n q_n^j,  w = sig(delta)*beta*gamma*0.25, q = 1-sig(delta)*sig(alpha)
//
// phase1: per-d 16x16 matrices T (intra-chunk causal Toeplitz), M (state collect),
//         P (carry inject), qc = q^16
// phase2: S = M x X per (d, 16-chunk tile)            [WMMA, TDM-staged X]
// phase3: exclusive scan H(c+1) = qc*H(c) + S(c)      [131072 lanes, float4 r/w]
// phase4: out = T x X + P x H + omega*x               [WMMA, TDM-staged X, b128 stores]
//
// Workspace floats: T/M/P (3*1024*256) + QC (1024*16) + S (1024*8*16*256) ~= 137.4 MB.

typedef __attribute__((ext_vector_type(2))) float v2f;
typedef __attribute__((ext_vector_type(8))) float v8f;
typedef __attribute__((ext_vector_type(4))) unsigned int v4u;
typedef __attribute__((ext_vector_type(4))) int v4i;
typedef __attribute__((ext_vector_type(8))) int v8i;

constexpr int Bz = 8;
constexpr int Lz = 4096;
constexpr int Dz = 1024;
constexpr int Nz = 16;
constexpr int NC = Lz / 16;     // 256 chunks of 16
constexpr int DT = 16;          // d's per block (one wave per d)

constexpr size_t OFF_T  = 0;
constexpr size_t OFF_M  = (size_t)Dz * 256;
constexpr size_t OFF_P  = 2 * (size_t)Dz * 256;
constexpr size_t OFF_QC = 3 * (size_t)Dz * 256;
constexpr size_t OFF_S  = OFF_QC + (size_t)Dz * Nz;

// ---- TDM: load a 16(d) x 256(t) f32 tile (row stride Dz) into LDS, padding
// ---- 1 DWORD per 16 DWORDs -> LDS row stride 17 (bank-conflict-free frags).
__device__ __forceinline__ void tdm_load_tile(const float* gsrc, void* lds_dst) {
  const unsigned lds_off = (unsigned)(uintptr_t)lds_dst;            // addr[31:0] = LDS offset
  const unsigned long long ga = (unsigned long long)(uintptr_t)gsrc;
  v4u g0;
  g0[0] = 1u;                                                       // count=1 (valid user D#)
  g0[1] = lds_off;                                                  // lds_addr
  g0[2] = (unsigned)(ga & 0xFFFFFFFFu);                             // global_addr[31:0]
  g0[3] = (unsigned)((ga >> 32) & 0x1FFFFFFu) | (2u << 30);         // global_addr[56:32], type=2
  v8i g1;
  g1[0] = (int)((2u << 16) | (1u << 20) | (3u << 22));  // data_size=4B, pad_en, pad per 16DW, pad 1DW
  g1[1] = (int)(((unsigned)Dz & 0xFFFFu) << 16);        // tensor_dim0 lo16 (=1024)
  g1[2] = (int)(((unsigned)Dz >> 16) | (((unsigned)(Bz * Lz) & 0xFFFFu) << 16)); // dim0 hi | dim1 lo
  g1[3] = (int)(((unsigned)(Bz * Lz) >> 16) | (16u << 16));         // dim1 hi | tile_dim0=16
  g1[4] = 256;                                                      // tile_dim1=256, tile_dim2=0
  g1[5] = Dz;                                                       // tensor_dim0_stride lo32
  g1[6] = 0;                                                        // stride0 hi | stride1 lo
  g1[7] = 0;                                                        // stride1 hi
  const v4i z4 = {0, 0, 0, 0};
#if defined(__clang_major__) && (__clang_major__ >= 23)
  const v8i z8 = {0, 0, 0, 0, 0, 0, 0, 0};
  __builtin_amdgcn_tensor_load_to_lds(g0, g1, z4, z4, z8, 0);
#else
  __builtin_amdgcn_tensor_load_to_lds(g0, g1, z4, z4, 0);
#endif
}

// ---------------- Phase 1: per-channel coefficient matrices ----------------
__global__ __launch_bounds__(256) void ema_setup(
    const float* __restrict__ delta, const float* __restrict__ alpha,
    const float* __restrict__ beta,  const float* __restrict__ gamma,
    float* __restrict__ ws) {
  const int d = blockIdx.x;
  const int tid = threadIdx.x;
  __shared__ float w_s[16];
  __shared__ float k_s[16];
  __shared__ float qp[16][17];   // qp[n][j] = q_n^j, j=0..16

  if (tid < 16) {
    const int n = tid;
    const float p = 1.f / (1.f + __expf(-delta[d * 16 + n]));
    const float a = 1.f / (1.f + __expf(-alpha[d * 16 + n]));
    const float q = 1.f - p * a;
    w_s[n] = p * beta[d * 16 + n] * gamma[d * 16 + n] * 0.25f;   // sqrt(1/16)
    float pw = 1.f;
    for (int j = 0; j <= 16; ++j) { qp[n][j] = pw; pw *= q; }
  }
  __syncthreads();
  if (tid < 16) {                           // k[j] = sum_n w_n q_n^j
    float s = 0.f;
    for (int n = 0; n < 16; ++n) s += w_s[n] * qp[n][tid];
    k_s[tid] = s;
  }
  __syncthreads();

  const int r = tid >> 4, c = tid & 15;
  ws[OFF_T + (size_t)d * 256 + tid] = (c <= r) ? k_s[r - c] : 0.f;  // T[tau][sig]
  ws[OFF_M + (size_t)d * 256 + tid] = qp[r][15 - c];                // M[n][sig]
  ws[OFF_P + (size_t)d * 256 + tid] = w_s[c] * qp[c][r + 1];        // P[tau][n]
  if (tid < 16) ws[OFF_QC + (size_t)d * 16 + tid] = qp[tid][16];
}

// ---------------- Phase 2: chunk states S = M x X (WMMA) ----------------
__global__ __launch_bounds__(512) void ema_states(
    const float* __restrict__ x, float* __restrict__ ws) {
  __shared__ float xs[256][17];            // filled by TDM (pad 1 DW / 16 DW)
  const int d0 = blockIdx.x * DT;
  const int b  = blockIdx.y;
  const int ct = blockIdx.z;               // tile of 16 chunks = 256 timesteps
  const int tid = threadIdx.x;

  const size_t gbase = ((size_t)b * Lz + (size_t)ct * 256) * Dz + d0;
  if (tid < 32) {                          // wave 0 issues one DMA for the block
    tdm_load_tile(x + gbase, &xs[0][0]);
    __builtin_amdgcn_s_wait_tensorcnt(0);
  }
  __syncthreads();

  const int w    = tid >> 5;               // wave -> channel d0+w
  const int lane = tid & 31;
  const int d    = d0 + w;
  const int row  = lane & 15;              // A: M-row alias B: N-col
  const int half = lane >> 4;

  const float* Mg = ws + OFF_M + (size_t)d * 256;
  v2f aM[4], bX[4];
#pragma unroll
  for (int kb = 0; kb < 4; ++kb) {
    const int k0 = 4 * kb + 2 * half;
    aM[kb].x = Mg[row * 16 + k0];
    aM[kb].y = Mg[row * 16 + k0 + 1];
    bX[kb].x = xs[row * 16 + k0][w];       // B[sigma][col]
    bX[kb].y = xs[row * 16 + k0 + 1][w];
  }
  v8f acc = {};
#pragma unroll
  for (int kb = 0; kb < 4; ++kb)
    acc = __builtin_amdgcn_wmma_f32_16x16x4_f32(false, aM[kb], false, bX[kb],
                                                (short)0, acc, false, false);
  // S layout [d][b][n][c]: half-wave rows 64B-contiguous over c.
  float* S = ws + OFF_S;
  const int c = ct * 16 + row;
#pragma unroll
  for (int r = 0; r < 8; ++r) {
    const int n = r + 8 * half;
    S[(((size_t)d * Bz + b) * Nz + n) * NC + c] = acc[r];
  }
}

// ---------------- Phase 3: exclusive scan over chunks (in place, float4) ----------------
__global__ __launch_bounds__(128) void ema_scan(float* __restrict__ ws) {
  const int d = blockIdx.x;
  const int tid = threadIdx.x;             // 128 = B*N lanes per d
  const int b = tid >> 4, n = tid & 15;
  float* base = ws + OFF_S + (((size_t)d * Bz + b) * Nz + n) * NC;
  const float qc = ws[OFF_QC + (size_t)d * 16 + n];
  float h = 0.f;
  for (int c4 = 0; c4 < NC / 4; ++c4) {
    float4 s = *(const float4*)(base + 4 * c4);
    float4 o;
    o.x = h; h = fmaf(qc, h, s.x);
    o.y = h; h = fmaf(qc, h, s.y);
    o.z = h; h = fmaf(qc, h, s.z);
    o.w = h; h = fmaf(qc, h, s.w);
    *(float4*)(base + 4 * c4) = o;
  }
}

// ---------------- Phase 4: out = T x X + P x H + omega * x (WMMA) ----------------
__global__ __launch_bounds__(512) void ema_output(
    const float* __restrict__ x, const float* __restrict__ omega,
    float* __restrict__ ws, float* __restrict__ out) {
  __shared__ float xs[256][17];
  const int d0 = blockIdx.x * DT;
  const int b  = blockIdx.y;
  const int ct = blockIdx.z;
  const int tid = threadIdx.x;

  const size_t gbase = ((size_t)b * Lz + (size_t)ct * 256) * Dz + d0;
  if (tid < 32) {
    tdm_load_tile(x + gbase, &xs[0][0]);
    __builtin_amdgcn_s_wait_tensorcnt(0);
  }
  __syncthreads();

  const int w    = tid >> 5;
  const int lane = tid & 31;
  const int d    = d0 + w;
  const int row  = lane & 15;
  const int half = lane >> 4;

  const float* Tg = ws + OFF_T + (size_t)d * 256;
  const float* Pg = ws + OFF_P + (size_t)d * 256;
  const float* H  = ws + OFF_S;
  const size_t hb = ((size_t)d * Bz + b) * Nz;

  v2f aT[4], aP[4], bX[4], bH[4];
#pragma unroll
  for (int kb = 0; kb < 4; ++kb) {
    const int k0 = 4 * kb + 2 * half;
    aT[kb].x = Tg[row * 16 + k0];
    aT[kb].y = Tg[row * 16 + k0 + 1];
    aP[kb].x = Pg[row * 16 + k0];
    aP[kb].y = Pg[row * 16 + k0 + 1];
    bX[kb].x = xs[row * 16 + k0][w];
    bX[kb].y = xs[row * 16 + k0 + 1][w];
    bH[kb].x = H[(hb + k0) * NC + ct * 16 + row];       // B[n][col]
    bH[kb].y = H[(hb + k0 + 1) * NC + ct * 16 + row];
  }
  v8f acc = {};
#pragma unroll
  for (int kb = 0; kb < 4; ++kb)
    acc = __builtin_amdgcn_wmma_f32_16x16x4_f32(false, aT[kb], false, bX[kb],
                                                (short)0, acc, false, false);
#pragma unroll
  for (int kb = 0; kb < 4; ++kb)
    acc = __builtin_amdgcn_wmma_f32_16x16x4_f32(false, aP[kb], false, bH[kb],
                                                (short)0, acc, false, false);

  // Residual + restage through LDS (each wave owns column w), then b128 stores.
  const float om = omega[d];
#pragma unroll
  for (int r = 0; r < 8; ++r) {
    const int tau = r + 8 * half;
    const int tt  = row * 16 + tau;        // t_local = chunk_col*16 + tau
    xs[tt][w] = acc[r] + om * xs[tt][w];
  }
  __syncthreads();
  for (int idx = tid; idx < 256 * 4; idx += 512) {
    const int tt = idx >> 2, i4 = (idx & 3) * 4;
    float4 v = make_float4(xs[tt][i4], xs[tt][i4 + 1], xs[tt][i4 + 2], xs[tt][i4 + 3]);
    *(float4*)(out + gbase + (size_t)tt * Dz + i4) = v;
  }
}

extern "C" void kernel_launch(void* const* d_in, const int* in_sizes, int n_in,
                              void* d_out, int out_size, void* d_ws, size_t ws_size,
                              hipStream_t stream) {
  (void)in_sizes; (void)n_in; (void)out_size; (void)ws_size;
  const float* x     = (const float*)d_in[0];
  const float* delta = (const float*)d_in[1];
  const float* alpha = (const float*)d_in[2];
  const float* beta  = (const float*)d_in[3];
  const float* gamma = (const float*)d_in[4];
  const float* omega = (const float*)d_in[5];
  float* ws  = (float*)d_ws;     // needs ~137.4 MB
  float* out = (float*)d_out;

  hipLaunchKernelGGL(ema_setup,  dim3(Dz), dim3(256), 0, stream,
                     delta, alpha, beta, gamma, ws);
  hipLaunchKernelGGL(ema_states, dim3(Dz / DT, Bz, NC / 16), dim3(512), 0, stream,
                     x, ws);
  hipLaunchKernelGGL(ema_scan,   dim3(Dz), dim3(128), 0, stream, ws);
  hipLaunchKernelGGL(ema_output, dim3(Dz / DT, Bz, NC / 16), dim3(512), 0, stream,
                     x, omega, ws, out);
}